// PRM_77824807403842
// MI455X (gfx1250) — compile-verified
//
#include <hip/hip_runtime.h>
#include <math.h>

// Problem dimensions (from setup_inputs): B=2,T=4,C=1 -> BT=8
#define BT  8
#define HS  270     // small height
#define WSM 480     // small width
#define HL  1080    // large height
#define WL  1920    // large width
#define BLK 160     // 5 wave32s; 3 blocks cover one 1920-pixel row (480 x4 cols)
#define PFROWS 8    // prefetch distance for the streaming 'large' read

typedef float         v4f  __attribute__((ext_vector_type(4)));
typedef unsigned char v4u8 __attribute__((ext_vector_type(4)));

__device__ __forceinline__ unsigned char trans_bit(float v, int sig) {
    const float LO = 1e-5f;
    const float HI = (float)(1.0 - 1e-5);   // match f32(1.0 - EPS)
    float t = sig ? (1.0f / (1.0f + expf(-v))) : v;
    return (unsigned char)((t > LO) && (t < HI));
}

// Pass 1: transition mask on the small (block) grid. 1,036,800 bytes total.
__global__ void PRM_mask_kernel(const float* __restrict__ small,
                                unsigned char* __restrict__ mask,
                                const int* __restrict__ sig_p, int n4) {
    int i = blockIdx.x * blockDim.x + threadIdx.x;
    if (i >= n4) return;
    int sig = *sig_p;
    v4f s = ((const v4f*)small)[i];
    v4u8 m;
    m.x = trans_bit(s.x, sig);
    m.y = trans_bit(s.y, sig);
    m.z = trans_bit(s.z, sig);
    m.w = trans_bit(s.w, sig);
    ((v4u8*)mask)[i] = m;
}

// Pass 2: fused upsample + dilation (collapsed to <=3x3 block-grid OR) + select.
// Grid: x = row segment (3 x 160 threads = 480 x4 cols), y = output row, z = image.
// One thread produces 4 contiguous output pixels (one small column p per thread).
template<bool FUSED>
__global__ void __launch_bounds__(BLK)
PRM_blend_kernel(const float* __restrict__ small,
                 const float* __restrict__ large,
                 const unsigned char* __restrict__ mask,
                 float* __restrict__ out,
                 const int* __restrict__ dil_p,
                 const int* __restrict__ sig_p) {
    const int x4 = blockIdx.x * BLK + threadIdx.x;  // small col / 16B output group
    const int y  = blockIdx.y;                      // uniform -> scalarized
    const int n  = blockIdx.z;                      // uniform
    const int q  = y >> 2;                          // small row (4x nearest)
    const int p  = x4;

    const int sig = FUSED ? *sig_p : 0;
    const float*         srow = small + (size_t)n * HS * WSM;
    const unsigned char* mrow = FUSED ? nullptr : (mask + (size_t)n * HS * WSM);

    auto mAt = [&](int qq, int pp) -> unsigned char {
        if (FUSED) return trans_bit(srow[qq * WSM + pp], sig);
        return mrow[qq * WSM + pp];
    };

    const int k = *dil_p;
    unsigned char c0m, c1m, c2m, c3m;

    if (k == 7) {
        // 7x7 window at 1080p == OR over <=3x3 block cells; clamped dup-safe.
        int r   = y & 3;                                  // uniform
        int rlo = q - (r == 3 ? 0 : 1); rlo = rlo < 0 ? 0 : rlo;          // uniform
        int rhi = q + (r == 0 ? 0 : 1); rhi = rhi > HS - 1 ? HS - 1 : rhi; // uniform
        int pl  = p - 1 < 0 ? 0 : p - 1;
        int ph  = p + 1 > WSM - 1 ? WSM - 1 : p + 1;
        unsigned char m0 = 0, m1 = 0, m2 = 0;
        for (int qq = rlo; qq <= rhi; ++qq) {             // uniform trip count
            m0 |= mAt(qq, pl);
            m1 |= mAt(qq, p);
            m2 |= mAt(qq, ph);
        }
        unsigned char a  = (unsigned char)(m0 | m1);   // cols p-1..p   (c=0)
        unsigned char b  = (unsigned char)(a  | m2);   // cols p-1..p+1 (c=1,2)
        unsigned char cc = (unsigned char)(m1 | m2);   // cols p..p+1   (c=3)
        c0m = a; c1m = b; c2m = b; c3m = cc;
    } else {
        // Generic width (k<1 -> identity). Padding is -inf => clamp-to-valid.
        int kk  = k < 1 ? 1 : k;
        int prL = (kk - 1) >> 1;
        int prR = kk - 1 - prL;
        int ylo = y - prL; ylo = ylo < 0 ? 0 : ylo;
        int yhi = y + prR; yhi = yhi > HL - 1 ? HL - 1 : yhi;
        int rlo = ylo >> 2, rhi = yhi >> 2;               // uniform
        unsigned char res[4];
        int xb = x4 << 2;
        for (int c = 0; c < 4; ++c) {
            int x   = xb + c;
            int xlo = x - prL; xlo = xlo < 0 ? 0 : xlo;
            int xhi = x + prR; xhi = xhi > WL - 1 ? WL - 1 : xhi;
            int clo = xlo >> 2, chi = xhi >> 2;
            unsigned char mm = 0;
            for (int qq = rlo; qq <= rhi; ++qq)
                for (int pp = clo; pp <= chi; ++pp)
                    mm |= mAt(qq, pp);
            res[c] = mm;
        }
        c0m = res[0]; c1m = res[1]; c2m = res[2]; c3m = res[3];
    }

    const float upv = srow[q * WSM + p];                  // cached (reused 4x)
    const size_t rowbase = (size_t)(n * HL + y) * WL;

    // Keep the HBM read stream deep: gfx1250 global_prefetch_b8 a few rows ahead.
    if (y + PFROWS < HL)
        __builtin_prefetch(large + rowbase + (size_t)PFROWS * WL + (x4 << 2), 0, 0);

    const v4f* lp = (const v4f*)(large + rowbase) + x4;   // streaming: NT
    v4f lg = __builtin_nontemporal_load(lp);

    v4f o;
    o.x = c0m ? lg.x : upv;   // exact: dilated trans in {0,1} => pure select
    o.y = c1m ? lg.y : upv;
    o.z = c2m ? lg.z : upv;
    o.w = c3m ? lg.w : upv;

    v4f* op = (v4f*)(out + rowbase) + x4;                 // streaming: NT
    __builtin_nontemporal_store(o, op);
}

extern "C" void kernel_launch(void* const* d_in, const int* in_sizes, int n_in,
                              void* d_out, int out_size, void* d_ws, size_t ws_size,
                              hipStream_t stream) {
    (void)in_sizes; (void)n_in; (void)out_size;
    const float* small = (const float*)d_in[0];
    const float* large = (const float*)d_in[1];
    const int*   dil   = (const int*)d_in[2];
    const int*   sig   = (const int*)d_in[3];
    float* out = (float*)d_out;

    const size_t MASK_BYTES = (size_t)BT * HS * WSM;      // ~1 MB (L2-resident)
    dim3 grid((WL / 4) / BLK, HL, BT);                    // (3, 1080, 8)
    dim3 block(BLK, 1, 1);

    if (d_ws != nullptr && ws_size >= MASK_BYTES) {
        unsigned char* msk = (unsigned char*)d_ws;
        const int n4 = (BT * HS * WSM) / 4;               // 259,200 (exact)
        PRM_mask_kernel<<<(n4 + 255) / 256, 256, 0, stream>>>(small, msk, sig, n4);
        PRM_blend_kernel<false><<<grid, block, 0, stream>>>(small, large, msk, out, dil, sig);
    } else {
        // Fallback: fuse mask computation into the blend kernel (no scratch).
        PRM_blend_kernel<true><<<grid, block, 0, stream>>>(small, large, nullptr, out, dil, sig);
    }
}